// SSDLoss_59382217834726
// MI455X (gfx1250) — compile-verified
//
#include <hip/hip_runtime.h>

#define NUM_A   98304
#define NUM_T   1024
#define NCLS    21
#define BLK     256
#define ABLKS   (NUM_A / BLK)   // 384, exact cover -> EXEC always full

typedef __attribute__((ext_vector_type(2))) float v2f;
typedef __attribute__((ext_vector_type(8))) float v8f;

__device__ __forceinline__ float focal_elem(float x, float tgt) {
    // sigmoid focal loss, alpha=0.25, gamma=2, numerically stable BCE
    float p   = 1.0f / (1.0f + expf(-x));
    float ce  = fmaxf(x, 0.0f) - x * tgt + log1pf(expf(-fabsf(x)));
    float p_t = p * tgt + (1.0f - p) * (1.0f - tgt);
    float a_t = 0.25f * tgt + 0.75f * (1.0f - tgt);
    float om  = 1.0f - p_t;
    return a_t * om * om * ce;
}

__device__ __forceinline__ float smooth_l1(float d) {
    float ad = fabsf(d);
    return (ad < 1.0f) ? 0.5f * d * d : ad - 0.5f;
}

// ---------------------------------------------------------------------------
// Kernel 1: per-anchor best target (argmax over T), fused focal loss,
// async global->LDS staging of the target boxes, WMMA wave sum reduction.
// ---------------------------------------------------------------------------
__global__ __launch_bounds__(BLK) void anchor_match_cls(
    const float* __restrict__ cls_preds,   // [A, 21]
    const float* __restrict__ anchors,     // [A, 4]
    const float* __restrict__ tboxes,      // [T, 4]
    const int*   __restrict__ tlabels,     // [T]
    float* __restrict__ cls_partial,       // [ABLKS]
    int*   __restrict__ pos_partial)       // [ABLKS]
{
    __shared__ float4 s_tb[NUM_T];             // 16 KB target boxes
    __shared__ float  s_ta[NUM_T];             // 4 KB target areas
    __shared__ float  s_red[(BLK / 32) * 16];  // per-wave WMMA column sums
    __shared__ int    s_cnt[BLK];

    const int tid = threadIdx.x;
    const float4* tb4 = (const float4*)tboxes;

    // CDNA5 async copy: target boxes go straight from global into LDS
    // (ASYNCcnt-tracked, no VGPR bounce). Flat addresses of __shared__ have
    // the LDS byte offset in the low 32 bits.
    for (int i = tid; i < NUM_T; i += BLK) {          // uniform trip count, no divergence
        unsigned lds_addr = (unsigned)(uintptr_t)(&s_tb[i]);
        unsigned long long ga = (unsigned long long)(uintptr_t)(tb4 + i);
        asm volatile("global_load_async_to_lds_b128 %0, %1, off"
                     :: "v"(lds_addr), "v"(ga) : "memory");
    }
#if defined(__has_builtin) && __has_builtin(__builtin_amdgcn_s_wait_asynccnt)
    __builtin_amdgcn_s_wait_asynccnt(0);
#else
    asm volatile("s_wait_asynccnt 0x0" ::: "memory");
#endif
    __syncthreads();
    for (int i = tid; i < NUM_T; i += BLK) {
        float4 b = s_tb[i];
        s_ta[i] = (b.z - b.x) * (b.w - b.y);
    }
    __syncthreads();

    const int aidx = blockIdx.x * BLK + tid;
    const float4 ab = ((const float4*)anchors)[aidx];
    const float area_a = (ab.z - ab.x) * (ab.w - ab.y);
    const float* cp = cls_preds + (size_t)aidx * NCLS;
    __builtin_prefetch(cp, 0, 0);   // global_prefetch_b8: warm cls row for after loop

    // argmax over targets; track (inter, union), compare via cross-products
    // (unions are strictly positive here). Strict '>' keeps first occurrence,
    // matching jnp.argmax semantics.
    float bi = 0.0f, bu = 1.0f;
    int   bt = 0;
#pragma unroll 4
    for (int t = 0; t < NUM_T; ++t) {
        float4 tb = s_tb[t];
        float lx = fmaxf(tb.x, ab.x);
        float ly = fmaxf(tb.y, ab.y);
        float rx = fminf(tb.z, ab.z);
        float ry = fminf(tb.w, ab.w);
        float w = fmaxf(rx - lx, 0.0f);
        float h = fmaxf(ry - ly, 0.0f);
        float inter = w * h;
        float uni = s_ta[t] + area_a - inter;
        bool gt = inter * bu > bi * uni;
        bi = gt ? inter : bi;
        bu = gt ? uni  : bu;
        bt = gt ? t    : bt;
    }
    float max_iou = bi / fmaxf(bu, 1e-8f);
    bool pos = max_iou >= 0.5f;
    int label = tlabels[bt];

    float fsum = 0.0f;
#pragma unroll
    for (int c = 0; c < NCLS; ++c) {
        float tgt = (pos && (c == label)) ? 1.0f : 0.0f;
        fsum += focal_elem(cp[c], tgt);
    }

    // Wave32 sum via V_WMMA_F32_16X16X4_F32 with A = all-ones:
    // D[0][n] = sum_k B[k][n]; every lane's b.x lands in exactly one B slot,
    // so sum over D row 0 (lanes 0..15, VGPR0) == sum over the 32 lane values.
    v2f a; a.x = 1.0f; a.y = 1.0f;
    v2f b; b.x = fsum; b.y = 0.0f;
    v8f cacc = {};
    cacc = __builtin_amdgcn_wmma_f32_16x16x4_f32(
        /*neg_a=*/false, a, /*neg_b=*/false, b,
        /*c_mod=*/(short)0, cacc, /*reuse_a=*/false, /*reuse_b=*/false);

    const int lane = tid & 31;
    const int wave = tid >> 5;
    if (lane < 16) s_red[wave * 16 + lane] = cacc[0];
    s_cnt[tid] = pos ? 1 : 0;
    __syncthreads();

    if (tid == 0) {
        float s = 0.0f;
        for (int i = 0; i < (BLK / 32) * 16; ++i) s += s_red[i];
        int np = 0;
        for (int i = 0; i < BLK; ++i) np += s_cnt[i];
        cls_partial[blockIdx.x] = s;
        pos_partial[blockIdx.x] = np;
    }
}

// ---------------------------------------------------------------------------
// Kernel 2: per-target best anchor (argmax over A), encode + smooth L1.
// One block per target.
// ---------------------------------------------------------------------------
__global__ __launch_bounds__(BLK) void target_match_reg(
    const float* __restrict__ boxes_preds, // [A, 4]
    const float* __restrict__ anchors,     // [A, 4]
    const float* __restrict__ tboxes,      // [T, 4]
    float* __restrict__ reg_partial,       // [T]
    int*   __restrict__ match_partial)     // [T]
{
    __shared__ float s_i[BLK];
    __shared__ float s_u[BLK];
    __shared__ int   s_x[BLK];

    const int t   = blockIdx.x;
    const int tid = threadIdx.x;
    const float4 tb = ((const float4*)tboxes)[t];
    const float area_t = (tb.z - tb.x) * (tb.w - tb.y);
    const float4* a4 = (const float4*)anchors;

    float bi = 0.0f, bu = 1.0f;
    int   bx = 0;
    for (int ai = tid; ai < NUM_A; ai += BLK) {
        float4 ab = a4[ai];
        float lx = fmaxf(tb.x, ab.x);
        float ly = fmaxf(tb.y, ab.y);
        float rx = fminf(tb.z, ab.z);
        float ry = fminf(tb.w, ab.w);
        float w = fmaxf(rx - lx, 0.0f);
        float h = fmaxf(ry - ly, 0.0f);
        float inter = w * h;
        float area_a = (ab.z - ab.x) * (ab.w - ab.y);
        float uni = area_t + area_a - inter;
        bool gt = inter * bu > bi * uni;   // strict > keeps lowest index
        bi = gt ? inter : bi;
        bu = gt ? uni  : bu;
        bx = gt ? ai   : bx;
    }
    s_i[tid] = bi; s_u[tid] = bu; s_x[tid] = bx;
    __syncthreads();

    for (int s = BLK / 2; s > 0; s >>= 1) {
        if (tid < s) {
            float i2 = s_i[tid + s], u2 = s_u[tid + s];
            int   x2 = s_x[tid + s];
            float l = s_i[tid] * u2;
            float r = i2 * s_u[tid];
            if (r > l || (r == l && x2 < s_x[tid])) {
                s_i[tid] = i2; s_u[tid] = u2; s_x[tid] = x2;
            }
        }
        __syncthreads();
    }

    if (tid == 0) {
        float iou = s_i[0] / fmaxf(s_u[0], 1e-8f);
        bool matched = iou >= 0.5f;
        float rsum = 0.0f;
        if (matched) {
            int mi = s_x[0];
            float4 ab = a4[mi];
            float aw = ab.z - ab.x, ah = ab.w - ab.y;
            float acx = ab.x + 0.5f * aw, acy = ab.y + 0.5f * ah;
            float bw = tb.z - tb.x, bh = tb.w - tb.y;
            float bcx = tb.x + 0.5f * bw, bcy = tb.y + 0.5f * bh;
            float e0 = (bcx - acx) / aw;
            float e1 = (bcy - acy) / ah;
            float e2 = logf(fmaxf(bw, 1e-8f) / aw);
            float e3 = logf(fmaxf(bh, 1e-8f) / ah);
            float4 pp = ((const float4*)boxes_preds)[mi];
            rsum = smooth_l1(pp.x - e0) + smooth_l1(pp.y - e1) +
                   smooth_l1(pp.z - e2) + smooth_l1(pp.w - e3);
        }
        reg_partial[t]   = rsum;
        match_partial[t] = matched ? 1 : 0;
    }
}

// ---------------------------------------------------------------------------
// Kernel 3: deterministic scalar finalize.
// ---------------------------------------------------------------------------
__global__ void finalize_loss(const float* __restrict__ cls_partial,
                              const int*   __restrict__ pos_partial,
                              const float* __restrict__ reg_partial,
                              const int*   __restrict__ match_partial,
                              float* __restrict__ out)
{
    if (threadIdx.x == 0 && blockIdx.x == 0) {
        float cs = 0.0f; int np = 0;
        for (int i = 0; i < ABLKS; ++i) { cs += cls_partial[i]; np += pos_partial[i]; }
        float rs = 0.0f; int nm = 0;
        for (int i = 0; i < NUM_T; ++i) { rs += reg_partial[i]; nm += match_partial[i]; }
        float cls_loss = cs / (float)np;                 // matches reference (inf if np==0)
        float nmf = fmaxf((float)nm, 1.0f);
        float reg_loss = rs / (nmf * 4.0f);
        out[0] = cls_loss + reg_loss;
        out[1] = cls_loss;
        out[2] = reg_loss;
    }
}

extern "C" void kernel_launch(void* const* d_in, const int* in_sizes, int n_in,
                              void* d_out, int out_size, void* d_ws, size_t ws_size,
                              hipStream_t stream) {
    const float* cls_preds   = (const float*)d_in[0];  // [A*21]
    const float* boxes_preds = (const float*)d_in[1];  // [A*4]
    const float* anchors     = (const float*)d_in[2];  // [A*4]
    const float* tboxes      = (const float*)d_in[3];  // [T*4]
    const int*   tlabels     = (const int*)d_in[4];    // [T]
    float* out = (float*)d_out;

    float* cls_partial   = (float*)d_ws;
    int*   pos_partial   = (int*)(cls_partial + ABLKS);
    float* reg_partial   = (float*)(pos_partial + ABLKS);
    int*   match_partial = (int*)(reg_partial + NUM_T);

    anchor_match_cls<<<ABLKS, BLK, 0, stream>>>(cls_preds, anchors, tboxes, tlabels,
                                                cls_partial, pos_partial);
    target_match_reg<<<NUM_T, BLK, 0, stream>>>(boxes_preds, anchors, tboxes,
                                                reg_partial, match_partial);
    finalize_loss<<<1, 32, 0, stream>>>(cls_partial, pos_partial,
                                        reg_partial, match_partial, out);
}